// RelativePositionEncoding_20779051778070
// MI455X (gfx1250) — compile-verified
//
#include <hip/hip_runtime.h>

// Problem constants (from the reference): x[4,4096,1024] f32, pe[65,1024] f32.
#define LQ   4096          // sequence length
#define DM   1024          // model dim
#define NB   4             // batch
#define MREL 32            // max relative position
#define NV   (2*MREL + 1)  // 65 buckets
#define COLS 256           // columns of agg handled per workgroup (DM/COLS = grid.y)
#define NTHREADS 256       // 8 wave32s

typedef __attribute__((ext_vector_type(2))) float v2f;
typedef __attribute__((ext_vector_type(4))) float v4f;
typedef __attribute__((ext_vector_type(8))) float v8f;

// Analytic histogram value counts[i][v] for clip(i-j, -m, m)+m over j in [0,L).
// Branchless selects so EXEC stays all-ones around the WMMA ops.
__device__ __forceinline__ float count_val(int i, int v) {
    int j = i - (v - MREL);                      // the single j hitting bucket v (middle case)
    float c = (j >= 0 && j < LQ) ? 1.0f : 0.0f;  // middle buckets 1..63
    int c_lo = LQ - MREL - i;                    // bucket 0  : #j >= i+32
    int c_hi = i - (MREL - 1);                   // bucket 64 : #j <= i-32
    c = (v == 0)        ? (float)(c_lo > 0 ? c_lo : 0) : c;
    c = (v == 2 * MREL) ? (float)(c_hi > 0 ? c_hi : 0) : c;
    c = (v >  2 * MREL) ? 0.0f : c;              // K-padding 65..67
    return c;
}

__global__ __launch_bounds__(NTHREADS) void relpos_fused(
    const float* __restrict__ x, const float* __restrict__ pe, float* __restrict__ out)
{
    __shared__ float agg_s[16 * COLS];           // 16 rows x 256 cols = 16 KB of the agg slice

    const int tid  = threadIdx.x;
    const int lane = tid & 31;
    const int wave = tid >> 5;                   // 0..7
    const int mrow = lane & 15;                  // M index (A) / N index (B, C/D)
    const int half = lane >> 4;                  // 0 or 1 (lane half)

    const int row0    = blockIdx.x * 16;         // 16 agg rows per WG
    const int colbase = blockIdx.y * COLS;       // 256 agg cols per WG
    const int i       = row0 + mrow;             // global query row this lane represents in A

    // ---- Phase 1: agg[16, COLS] = (counts @ pe) / L via V_WMMA_F32_16X16X4_F32 ----
    // A-fragments (counts) do not depend on the column tile: build all 17 K-steps once.
    // 32-bit A 16x4 layout: lanes 0-15 -> {K=k0, k0+1}, lanes 16-31 -> {K=k0+2, k0+3}.
    v2f afrag[17];
    #pragma unroll
    for (int s = 0; s < 17; ++s) {
        int ka = s * 4 + half * 2;
        afrag[s].x = count_val(i, ka);
        afrag[s].y = count_val(i, ka + 1);
    }

    const float scale = 1.0f / (float)LQ;

    #pragma unroll
    for (int t = 0; t < COLS / (16 * 8); ++t) {          // 2 column tiles per wave
        const int col_local = (wave * (COLS / (16 * 8)) + t) * 16;
        const int col       = colbase + col_local + mrow; // B: lane -> column N
        v8f acc = {};
        #pragma unroll
        for (int s = 0; s < 17; ++s) {                   // K = 0..67 (65..67 zero-padded)
            int kb = s * 4 + half * 2;                   // B 4x16 layout mirrors A
            v2f b;
            b.x = (kb     < NV) ? pe[(size_t)kb       * DM + col] : 0.0f;
            b.y = (kb + 1 < NV) ? pe[(size_t)(kb + 1) * DM + col] : 0.0f;
            acc = __builtin_amdgcn_wmma_f32_16x16x4_f32(
                      /*neg_a=*/false, afrag[s], /*neg_b=*/false, b,
                      /*c_mod=*/(short)0, acc, /*reuse_a=*/false, /*reuse_b=*/false);
        }
        // C/D 16x16 f32 layout: VGPR r -> row (r + 8*half), lane -> col.
        #pragma unroll
        for (int r = 0; r < 8; ++r) {
            agg_s[(r + half * 8) * COLS + col_local + mrow] = acc[r] * scale;
        }
    }

    __syncthreads();

    // ---- Phase 2: out = x + agg (broadcast over batch), pure streaming ----
    // 128-bit non-temporal loads/stores: x/out are touched once, keep pe hot in L2.
    const v4f* __restrict__ x4 = reinterpret_cast<const v4f*>(x);
    v4f*       __restrict__ o4 = reinterpret_cast<v4f*>(out);
    const v4f* __restrict__ a4 = reinterpret_cast<const v4f*>(agg_s);

    #pragma unroll 4
    for (int it = 0; it < (NB * 16 * COLS / 4) / NTHREADS; ++it) {  // 16 iterations
        int idx = it * NTHREADS + tid;
        int d4  = idx & (COLS / 4 - 1);          // float4 column within slice (coalesced)
        int rr  = (idx >> 6) & 15;               // row within the 16-row slice
        int b   = idx >> 10;                     // batch
        size_t g4 = (size_t)b * (LQ * (DM / 4))
                  + (size_t)(row0 + rr) * (DM / 4)
                  + (colbase >> 2) + d4;
        v4f xv = __builtin_nontemporal_load(x4 + g4);
        v4f av = a4[rr * (COLS / 4) + d4];
        __builtin_nontemporal_store(xv + av, o4 + g4);
    }
}

extern "C" void kernel_launch(void* const* d_in, const int* in_sizes, int n_in,
                              void* d_out, int out_size, void* d_ws, size_t ws_size,
                              hipStream_t stream) {
    const float* x  = (const float*)d_in[0];   // [4, 4096, 1024] f32
    const float* pe = (const float*)d_in[1];   // [65, 1024] f32
    float* out = (float*)d_out;                // [4, 4096, 1024] f32
    (void)in_sizes; (void)n_in; (void)out_size; (void)d_ws; (void)ws_size;

    dim3 grid(LQ / 16, DM / COLS);             // (256, 4) = 1024 workgroups
    relpos_fused<<<grid, NTHREADS, 0, stream>>>(x, pe, out);
}